// CustomModel_30451318129113
// MI455X (gfx1250) — compile-verified
//
#include <hip/hip_runtime.h>
#include <math.h>

// ---------------------------------------------------------------------------
// CDNA5 (gfx1250) implementation of the GCLSTM reference.
//
// Key algebraic facts exploited (exact, not approximations):
//   * H = C0 = 0 entering every GCLSTM cell, so every ChebConv contributes
//     only its bias cb_g, the forget gate is multiplied by 0, and wc_i/wc_f
//     peepholes vanish.  edge_weight / cheb-norm are never needed.
//   * Remaining per stage: 3 GEMMs (W_i, W_c, W_o) + gate math -> H,
//     GEMM H@conv_W, GCN scatter (A+I sym-normalized), ReLU, BatchNorm,
//     pairwise channel maxpool.
//
// All GEMMs use V_WMMA_F32_16X16X4_F32 (fp32 tensor op, no precision loss).
// N = 50000 is divisible by 16 and all channel counts by 16, so WMMA tiles
// need no bounds checks (EXEC stays all-ones as the ISA requires).
// ---------------------------------------------------------------------------

typedef float v2f __attribute__((ext_vector_type(2)));
typedef float v8f __attribute__((ext_vector_type(8)));

#define N_NODES 50000
#define N_EDGES 800000

__device__ __forceinline__ float sigm_(float x) { return 1.0f / (1.0f + __expf(-x)); }

// --------------------------------------------------------------------------
// Gate kernel: for a 16x16 output tile, accumulate X@W_i, X@W_c, X@W_o with
// fp32 WMMA, then fuse the LSTM gate nonlinearity into H.
// Block = 1 wave (32 lanes). grid = (N/16, cout/16).
//
// fp32 WMMA fragment layouts (ISA 7.12.2):
//   A(16x4):  lane l<16 holds M=l, K={0,1}; lane l>=16 holds M=l-16, K={2,3}
//   B(4x16):  VGPR v, lane l: K = v + 2*(l>=16), N = l&15
//   C/D:      VGPR v, lane l: M = v + 8*(l>=16), N = l&15
// --------------------------------------------------------------------------
__global__ __launch_bounds__(32) void gates_wmma_kernel(
    const float* __restrict__ X, int cin, int cout,
    const float* __restrict__ Wi, const float* __restrict__ Wc,
    const float* __restrict__ Wo,
    const float* __restrict__ bi, const float* __restrict__ bc,
    const float* __restrict__ bo,
    const float* __restrict__ cbi, const float* __restrict__ cbc,
    const float* __restrict__ cbo,
    const float* __restrict__ wco,
    float* __restrict__ H)
{
    const int lane = threadIdx.x;
    const int half = lane >> 4;   // 0: lanes 0-15, 1: lanes 16-31
    const int l16  = lane & 15;
    const int m0 = blockIdx.x * 16;
    const int c0 = blockIdx.y * 16;

    v8f accI = {}; v8f accC = {}; v8f accO = {};

    const int rowA = m0 + l16;
    for (int k0 = 0; k0 < cin; k0 += 4) {
        const int ka = k0 + 2 * half;
        v2f a;
        a.x = X[(size_t)rowA * cin + ka + 0];
        a.y = X[(size_t)rowA * cin + ka + 1];

        const size_t b0 = (size_t)(ka + 0) * cout + c0 + l16;
        const size_t b1 = (size_t)(ka + 1) * cout + c0 + l16;
        v2f bI, bC, bO;
        bI.x = Wi[b0]; bI.y = Wi[b1];
        bC.x = Wc[b0]; bC.y = Wc[b1];
        bO.x = Wo[b0]; bO.y = Wo[b1];

        accI = __builtin_amdgcn_wmma_f32_16x16x4_f32(false, a, false, bI, (short)0, accI, false, false);
        accC = __builtin_amdgcn_wmma_f32_16x16x4_f32(false, a, false, bC, (short)0, accC, false, false);
        accO = __builtin_amdgcn_wmma_f32_16x16x4_f32(false, a, false, bO, (short)0, accO, false, false);
    }

    const int col = c0 + l16;
    const float Bi = bi[col] + cbi[col];   // ChebConv(H=0) == its bias
    const float Bc = bc[col] + cbc[col];
    const float Bo = bo[col] + cbo[col];
    const float Wpeep = wco[col];

#pragma unroll
    for (int v = 0; v < 8; ++v) {
        const int row = m0 + v + 8 * half;
        const float Iv = sigm_(accI[v] + Bi);
        const float Tv = tanhf(accC[v] + Bc);
        const float Cv = Iv * Tv;                       // F*C0 == 0
        const float Ov = sigm_(accO[v] + Bo + Wpeep * Cv);
        H[(size_t)row * cout + col] = Ov * tanhf(Cv);
    }
}

// Plain fp32 WMMA GEMM: Y[M x Nc] = A[M x K] @ W[K x Nc]; grid (M/16, Nc/16).
__global__ __launch_bounds__(32) void gemm_wmma_kernel(
    const float* __restrict__ A, int K, int Nc,
    const float* __restrict__ W, float* __restrict__ Y)
{
    const int lane = threadIdx.x;
    const int half = lane >> 4;
    const int l16  = lane & 15;
    const int m0 = blockIdx.x * 16;
    const int c0 = blockIdx.y * 16;

    v8f acc = {};
    const int rowA = m0 + l16;
    for (int k0 = 0; k0 < K; k0 += 4) {
        const int ka = k0 + 2 * half;
        v2f a, b;
        a.x = A[(size_t)rowA * K + ka + 0];
        a.y = A[(size_t)rowA * K + ka + 1];
        b.x = W[(size_t)(ka + 0) * Nc + c0 + l16];
        b.y = W[(size_t)(ka + 1) * Nc + c0 + l16];
        acc = __builtin_amdgcn_wmma_f32_16x16x4_f32(false, a, false, b, (short)0, acc, false, false);
    }
#pragma unroll
    for (int v = 0; v < 8; ++v) {
        const int row = m0 + v + 8 * half;
        Y[(size_t)row * Nc + c0 + l16] = acc[v];
    }
}

// ---- GCN degree (A + I, counted over destination col, incl. self loop) ----
__global__ void deg_init(float* __restrict__ dinv) {
    int v = blockIdx.x * blockDim.x + threadIdx.x;
    if (v < N_NODES) dinv[v] = 1.0f;                  // self loop
}
__global__ void deg_accum(const int* __restrict__ cols, float* __restrict__ dinv) {
    int e = blockIdx.x * blockDim.x + threadIdx.x;
    if (e < N_EDGES) atomicAdd(&dinv[cols[e]], 1.0f);
}
__global__ void deg_fin(float* __restrict__ dinv) {
    int v = blockIdx.x * blockDim.x + threadIdx.x;
    if (v < N_NODES) dinv[v] = rsqrtf(dinv[v]);       // deg >= 1 always
}

// OUT = conv_b + dinv[v]^2 * XW   (self-loop contribution, non-atomic init)
__global__ void out_init_kernel(const float* __restrict__ XW,
                                const float* __restrict__ dinv,
                                const float* __restrict__ convb,
                                float* __restrict__ OUT, int cout) {
    long i = (long)blockIdx.x * blockDim.x + threadIdx.x;
    if (i >= (long)N_NODES * cout) return;
    int v = (int)(i / cout), ch = (int)(i % cout);
    float dv = dinv[v];
    OUT[i] = convb[ch] + dv * dv * XW[i];
}

// Edge scatter: OUT[col] += d[row]*d[col] * XW[row]   (atomic f32 adds)
__global__ void scatter_kernel(const int* __restrict__ rows,
                               const int* __restrict__ cols,
                               const float* __restrict__ dinv,
                               const float* __restrict__ XW,
                               float* __restrict__ OUT, int cout) {
    int e = blockIdx.x * blockDim.x + threadIdx.x;
    if (e >= N_EDGES) return;
    const int r = rows[e], c = cols[e];
    const float nrm = dinv[r] * dinv[c];
    const float4* __restrict__ src = (const float4*)(XW + (size_t)r * cout);
    float* __restrict__ dst = OUT + (size_t)c * cout;
    __builtin_prefetch(dst, 1, 3);                    // global_prefetch_b8
    const int q = cout >> 2;
    for (int j = 0; j < q; ++j) {
        float4 x = src[j];
        atomicAdd(dst + 4 * j + 0, nrm * x.x);
        atomicAdd(dst + 4 * j + 1, nrm * x.y);
        atomicAdd(dst + 4 * j + 2, nrm * x.z);
        atomicAdd(dst + 4 * j + 3, nrm * x.w);
    }
}

__global__ void zero_stats(float* __restrict__ s, int n) {
    int i = blockIdx.x * blockDim.x + threadIdx.x;
    if (i < n) s[i] = 0.0f;
}

// Column sums of relu(OUT) and relu(OUT)^2; blockDim.x == cout, one column
// per thread over a row chunk, then one atomic per column per block.
__global__ void bn_stats_kernel(const float* __restrict__ OUT,
                                float* __restrict__ stats, int cout,
                                int rows_per_block) {
    const int col = threadIdx.x;
    int r0 = blockIdx.x * rows_per_block;
    int r1 = r0 + rows_per_block; if (r1 > N_NODES) r1 = N_NODES;
    float s = 0.0f, s2 = 0.0f;
    for (int r = r0; r < r1; ++r) {
        float x = OUT[(size_t)r * cout + col];
        x = fmaxf(x, 0.0f);
        s += x; s2 += x * x;
    }
    atomicAdd(&stats[col], s);
    atomicAdd(&stats[cout + col], s2);
}

// BatchNorm (population var, eps=1e-5) on relu(OUT), then MaxPool1d(2) over
// channel pairs -> next stage input.
__global__ void bn_pool_kernel(const float* __restrict__ OUT,
                               const float* __restrict__ stats,
                               const float* __restrict__ gamma,
                               const float* __restrict__ beta,
                               float* __restrict__ Xn, int cout) {
    const int halfc = cout >> 1;
    long i = (long)blockIdx.x * blockDim.x + threadIdx.x;
    if (i >= (long)N_NODES * halfc) return;
    int v = (int)(i / halfc), j = (int)(i % halfc);
    const float invN = 1.0f / (float)N_NODES;
    float y[2];
#pragma unroll
    for (int t = 0; t < 2; ++t) {
        int ch = 2 * j + t;
        float mu  = stats[ch] * invN;
        float var = stats[cout + ch] * invN - mu * mu;
        float r = fmaxf(OUT[(size_t)v * cout + ch], 0.0f);
        y[t] = gamma[ch] * (r - mu) * rsqrtf(var + 1e-5f) + beta[ch];
    }
    Xn[(size_t)v * halfc + j] = fmaxf(y[0], y[1]);
}

__global__ void linear_kernel(const float* __restrict__ X8,
                              const float* __restrict__ linW,
                              const float* __restrict__ linb,
                              float* __restrict__ out) {
    int v = blockIdx.x * blockDim.x + threadIdx.x;
    if (v >= N_NODES) return;
    float s = linb[0];
#pragma unroll
    for (int k = 0; k < 8; ++k) s += X8[(size_t)v * 8 + k] * linW[k];
    out[v] = s;
}

// ---------------------------------------------------------------------------
// Host side.
//
// Input index map (top-level dict insertion order; nested `params` assumed to
// flatten as a JAX pytree, i.e. keys sorted ASCIIbetically — capitals first):
//   d_in[0] x (50000*32 f32)   d_in[1] edge_index (2*800000 i32, row-major)
//   d_in[2] edge_weight (unused)  d_in[3] lin_W (8)  d_in[4] lin_b (1)
//   d_in[5 + 23*s + leaf] for stage s in {0..3}, leaf order:
//     T_c T_f T_i T_o  W_c W_f W_i W_o  b_c b_f b_i b_o  beta
//     cb_c cb_f cb_i cb_o  conv_W conv_b  gamma  wc_f wc_i wc_o
// ---------------------------------------------------------------------------
enum {
    L_T_c = 0, L_T_f, L_T_i, L_T_o,
    L_W_c, L_W_f, L_W_i, L_W_o,
    L_b_c, L_b_f, L_b_i, L_b_o,
    L_beta,
    L_cb_c, L_cb_f, L_cb_i, L_cb_o,
    L_conv_W, L_conv_b, L_gamma,
    L_wc_f, L_wc_i, L_wc_o
};

extern "C" void kernel_launch(void* const* d_in, const int* in_sizes, int n_in,
                              void* d_out, int out_size, void* d_ws, size_t ws_size,
                              hipStream_t stream) {
    (void)in_sizes; (void)n_in; (void)out_size; (void)ws_size;

    const float* x    = (const float*)d_in[0];
    const int*   ei   = (const int*)d_in[1];
    const int*   erow = ei;
    const int*   ecol = ei + N_EDGES;
    const float* linW = (const float*)d_in[3];
    const float* linb = (const float*)d_in[4];
    auto sp = [&](int stage, int leaf) -> const float* {
        return (const float*)d_in[5 + 23 * stage + leaf];
    };

    // Workspace: three N x 128 fp32 buffers + degree + BN stats.
    char* ws = (char*)d_ws;
    const size_t big = (size_t)N_NODES * 128 * sizeof(float);   // 25.6 MB
    float* BUF0  = (float*)(ws);                 // H, then OUT (reused)
    float* BUF1  = (float*)(ws + big);           // XW
    float* BUF2  = (float*)(ws + 2 * big);       // pooled X for next stage
    float* dinv  = (float*)(ws + 3 * big);       // N floats
    float* stats = (float*)(ws + 3 * big + ((size_t)N_NODES * 4 + 255) / 256 * 256);

    // GCN symmetric norm degrees (same for all stages; cheap, recomputed
    // every call so the launch stays deterministic/stateless).
    deg_init <<<(N_NODES + 255) / 256, 256, 0, stream>>>(dinv);
    deg_accum<<<(N_EDGES + 255) / 256, 256, 0, stream>>>(ecol, dinv);
    deg_fin  <<<(N_NODES + 255) / 256, 256, 0, stream>>>(dinv);

    const int cins[4]  = {32, 64, 32, 16};
    const int couts[4] = {128, 64, 32, 16};
    const float* Xsrc = x;

    for (int s = 0; s < 4; ++s) {
        const int cin = cins[s], cout = couts[s];

        // H = GCLSTM gates (3 WMMA GEMMs + fused nonlinearity)
        gates_wmma_kernel<<<dim3(N_NODES / 16, cout / 16), 32, 0, stream>>>(
            Xsrc, cin, cout,
            sp(s, L_W_i), sp(s, L_W_c), sp(s, L_W_o),
            sp(s, L_b_i), sp(s, L_b_c), sp(s, L_b_o),
            sp(s, L_cb_i), sp(s, L_cb_c), sp(s, L_cb_o),
            sp(s, L_wc_o), BUF0);

        // XW = H @ conv_W
        gemm_wmma_kernel<<<dim3(N_NODES / 16, cout / 16), 32, 0, stream>>>(
            BUF0, cout, cout, sp(s, L_conv_W), BUF1);

        // OUT = conv_b + self-loop term, then edge scatter (atomics)
        long nelem = (long)N_NODES * cout;
        out_init_kernel<<<(unsigned)((nelem + 255) / 256), 256, 0, stream>>>(
            BUF1, dinv, sp(s, L_conv_b), BUF0, cout);
        scatter_kernel<<<(N_EDGES + 255) / 256, 256, 0, stream>>>(
            erow, ecol, dinv, BUF1, BUF0, cout);

        // BatchNorm stats over relu(OUT), then normalize + maxpool(2)
        zero_stats<<<1, 256, 0, stream>>>(stats, 2 * cout);
        bn_stats_kernel<<<(N_NODES + 127) / 128, cout, 0, stream>>>(
            BUF0, stats, cout, 128);
        long npool = (long)N_NODES * (cout / 2);
        bn_pool_kernel<<<(unsigned)((npool + 255) / 256), 256, 0, stream>>>(
            BUF0, stats, sp(s, L_gamma), sp(s, L_beta), BUF2, cout);

        Xsrc = BUF2;
    }

    linear_kernel<<<(N_NODES + 255) / 256, 256, 0, stream>>>(
        BUF2, linW, linb, (float*)d_out);
}